// OneWinnerMHN_4879082848727
// MI455X (gfx1250) — compile-verified
//
#include <hip/hip_runtime.h>
#include <hip/hip_bf16.h>

typedef __attribute__((ext_vector_type(2))) float v2f;
typedef __attribute__((ext_vector_type(8))) float v8f;

#define BATCHN 128
#define SEQ    64
#define INPUT  320
#define ITEM   256
#define OUTD   64
#define HID    1024
#define OH     192
#define NPAD   80     // 64 ctx cols + 1 query col + 15 pad
#define NT     5      // N tiles of 16
#define MROWS  128    // rows per block = 8 waves * 16
#define KC     16     // K chunk (4 WMMA k-steps of 4)
#define ASTR   18     // A LDS row stride (even -> 8B-aligned pair loads, conflict-free)
#define BSTR   84     // B LDS row stride in float2 (k-pair interleaved)

// ---------------------------------------------------------------------------
// Kernel 1: batched GEMM with fp32 WMMA.
//   blockIdx.x in [0,7]: L0[b][mb*128 .. +127][0..79] = W_hi rows x [Xitem^T|x_q]
//   blockIdx.x == 8    : G[b][0..63][0..79]           = Xitem rows x [Xitem^T|x_q]
// A fragment (16x4 f32): lane L -> M = L&15, K-pair base = 4s + 2*(L>>4).
// B fragment dual layout; C tile: vgpr v -> M = v + 8*(L>>4), N = L&15.
// ---------------------------------------------------------------------------
__global__ __launch_bounds__(256)
void k_gemm(const float* __restrict__ Whi,   // [B][HID][ITEM]
            const float* __restrict__ xc,    // [B][SEQ][INPUT]
            const float* __restrict__ xq,    // [B][ITEM]
            float* __restrict__ L0,          // [B][HID][NPAD]
            float* __restrict__ G)           // [B][SEQ][NPAD]
{
    const int b    = blockIdx.y;
    const int mb   = blockIdx.x;
    const bool gram = (mb == 8);
    const int tid  = threadIdx.x;
    const int wave = tid >> 5;
    const int lane = tid & 31;

    __shared__ float a_s[MROWS * ASTR];        // 128 rows x 16 k (stride 18)
    __shared__ v2f   b2_s[(KC / 2) * BSTR];    // 8 k-pairs x 80 n (+4 pad), float2 each

    const float* Abase;
    int lda, arows;
    if (!gram) {
        Abase = Whi + ((size_t)b * HID + (size_t)mb * MROWS) * ITEM;
        lda = ITEM; arows = MROWS;
    } else {
        Abase = xc + (size_t)b * SEQ * INPUT;   // item part = first 256 of each row
        lda = INPUT; arows = SEQ;
    }
    const float* Bc = xc + (size_t)b * SEQ * INPUT;
    const float* Bq = xq + (size_t)b * ITEM;

    // zero pad columns 65..79 once (never rewritten inside the K loop)
    for (int i = tid; i < (KC / 2) * 15; i += 256) {
        int k2 = i / 15, c = 65 + (i % 15);
        v2f z; z[0] = 0.0f; z[1] = 0.0f;
        b2_s[k2 * BSTR + c] = z;
    }

    v8f acc[NT];
    #pragma unroll
    for (int n = 0; n < NT; ++n)
        #pragma unroll
        for (int e = 0; e < 8; ++e) acc[n][e] = 0.0f;

    for (int k0 = 0; k0 < ITEM; k0 += KC) {
        __syncthreads();   // previous iteration's LDS reads done before restaging
        // ---- stage A chunk: 128 rows x 16 k (coalesced float4 loads) ----
        {
            int row = tid >> 1;
            int off = (tid & 1) * 8;
            float4 r0 = {0.f, 0.f, 0.f, 0.f}, r1 = {0.f, 0.f, 0.f, 0.f};
            if (row < arows) {
                const float* p = Abase + (size_t)row * lda + k0 + off;
                r0 = *(const float4*)(p);
                r1 = *(const float4*)(p + 4);
                if (k0 + KC < ITEM)  // warm next chunk (global_prefetch)
                    __builtin_prefetch(p + KC, 0, 1);
            }
            v2f* d = (v2f*)(a_s + row * ASTR + off);   // 8B aligned (even index)
            v2f t0; t0[0] = r0.x; t0[1] = r0.y;
            v2f t1; t1[0] = r0.z; t1[1] = r0.w;
            v2f t2; t2[0] = r1.x; t2[1] = r1.y;
            v2f t3; t3[0] = r1.z; t3[1] = r1.w;
            d[0] = t0; d[1] = t1; d[2] = t2; d[3] = t3;
        }
        // ---- stage B chunk, K-pair interleaved: b2[k/2][n] = {B[k][n], B[k+1][n]} ----
        {
            int n   = tid >> 2;
            int seg = (tid & 3) * 4;          // k offset within chunk: 0,4,8,12
            int k2  = seg >> 1;               // k-pair row: 0,2,4,6
            float4 r = *(const float4*)(Bc + (size_t)n * INPUT + k0 + seg);
            v2f lo; lo[0] = r.x; lo[1] = r.y;
            v2f hi; hi[0] = r.z; hi[1] = r.w;
            b2_s[k2 * BSTR + n]       = lo;
            b2_s[(k2 + 1) * BSTR + n] = hi;
            if (tid < 4) {
                int sg = tid * 4;
                float4 q = *(const float4*)(Bq + k0 + sg);
                v2f ql; ql[0] = q.x; ql[1] = q.y;
                v2f qh; qh[0] = q.z; qh[1] = q.w;
                b2_s[(sg >> 1) * BSTR + 64]       = ql;
                b2_s[((sg >> 1) + 1) * BSTR + 64] = qh;
            }
        }
        __syncthreads();
        // ---- 4 WMMA k-steps over the chunk: 1 b64 LDS load per fragment ----
        const int mrow = wave * 16 + (lane & 15);
        const int kh   = lane >> 4;           // 0 or 1: which K-pair of the step
        #pragma unroll
        for (int s = 0; s < 4; ++s) {
            v2f afrag = *(const v2f*)(a_s + mrow * ASTR + s * 4 + kh * 2);
            const int k2 = s * 2 + kh;
            #pragma unroll
            for (int n = 0; n < NT; ++n) {
                v2f bfrag = b2_s[k2 * BSTR + n * 16 + (lane & 15)];
                acc[n] = __builtin_amdgcn_wmma_f32_16x16x4_f32(
                    false, afrag, false, bfrag, (short)0, acc[n], false, false);
            }
        }
    }

    if (gram && wave >= 4) return;   // gram block only has 64 valid rows
    float* Obase = gram ? (G + (size_t)b * SEQ * NPAD)
                        : (L0 + ((size_t)b * HID + (size_t)mb * MROWS) * NPAD);
    const int rbase = wave * 16 + (lane >> 4) * 8;
    const int col0  = lane & 15;
    #pragma unroll
    for (int n = 0; n < NT; ++n)
        #pragma unroll
        for (int v = 0; v < 8; ++v)
            Obase[(size_t)(rbase + v) * NPAD + n * 16 + col0] = acc[n][v];
}

// ---------------------------------------------------------------------------
// Kernel 2: sequential one-winner scan + softmax + output matvecs.
// One block (256 threads) per batch; each thread owns h = tid*4..tid*4+3.
// ---------------------------------------------------------------------------
__global__ __launch_bounds__(256)
void k_scan(const float* __restrict__ L0,   // [B][HID][NPAD]
            const float* __restrict__ Gws,  // [B][SEQ][NPAD]
            const float* __restrict__ xc,   // [B][SEQ][INPUT]
            const float* __restrict__ ctx,  // [B][SEQ][OH]
            const float* __restrict__ Woh,  // [B][OUTD][HID]
            const float* __restrict__ Wre,  // [B][HID][OH]
            float* __restrict__ outp)       // [B*OUTD] then [B*OH]
{
    const int b    = blockIdx.x;
    const int tid  = threadIdx.x;
    const int wave = tid >> 5;
    const int lane = tid & 31;

    __shared__ float Gs[SEQ * 65];   // G rows 0..63, cols 0..64 (col64 = gq)
    __shared__ int   lw[HID];        // last-writer step per hidden row, -1 = none
    __shared__ float av[HID];        // logits -> exp -> masked softmax
    __shared__ float bvec[SEQ];      // a[h] routed to its writing step
    __shared__ float labp[OUTD];
    __shared__ float redv[8];
    __shared__ int   redi[8];
    __shared__ float rscal[8];

    const float* Gb = Gws + (size_t)b * SEQ * NPAD;
    for (int i = tid; i < SEQ * 65; i += 256) {
        int r = i / 65, c = i - r * 65;
        Gs[i] = Gb[r * NPAD + c];
    }
    for (int i = tid; i < HID; i += 256) lw[i] = -1;
    if (tid < SEQ) bvec[tid] = 0.0f;
    __syncthreads();

    const float* Lb = L0 + (size_t)b * HID * NPAD;

    // ---- sequential winner scan ----
    for (int t = 0; t < SEQ; ++t) {
        float bv = -INFINITY; int bi = 0x7fffffff;
        #pragma unroll
        for (int j = 0; j < 4; ++j) {
            int h = tid * 4 + j;
            int s = lw[h];
            float v = (s < 0) ? Lb[(size_t)h * NPAD + t] : Gs[s * 65 + t];
            if (v > bv) { bv = v; bi = h; }           // keeps first max (low h)
        }
        #pragma unroll
        for (int o = 16; o > 0; o >>= 1) {
            float ov = __shfl_down(bv, o, 32);
            int   oi = __shfl_down(bi, o, 32);
            if (ov > bv || (ov == bv && oi < bi)) { bv = ov; bi = oi; }
        }
        if (lane == 0) { redv[wave] = bv; redi[wave] = bi; }
        __syncthreads();
        if (tid == 0) {
            float mv = redv[0]; int mi = redi[0];
            for (int w = 1; w < 8; ++w)
                if (redv[w] > mv || (redv[w] == mv && redi[w] < mi)) { mv = redv[w]; mi = redi[w]; }
            lw[mi] = t;
        }
        __syncthreads();
    }

    // ---- query logits under final W_hi, then softmax over 1024 ----
    float lmax = -INFINITY;
    #pragma unroll
    for (int j = 0; j < 4; ++j) {
        int h = tid * 4 + j;
        int s = lw[h];
        float v = (s < 0) ? Lb[(size_t)h * NPAD + SEQ] : Gs[s * 65 + SEQ];
        av[h] = v;
        lmax = fmaxf(lmax, v);
    }
    #pragma unroll
    for (int o = 16; o > 0; o >>= 1) lmax = fmaxf(lmax, __shfl_down(lmax, o, 32));
    if (lane == 0) redv[wave] = lmax;
    __syncthreads();
    if (tid == 0) {
        float m = redv[0];
        for (int w = 1; w < 8; ++w) m = fmaxf(m, redv[w]);
        redv[0] = m;
    }
    __syncthreads();
    lmax = redv[0];
    float lsum = 0.f;
    #pragma unroll
    for (int j = 0; j < 4; ++j) {
        int h = tid * 4 + j;
        float e = expf(av[h] - lmax);
        av[h] = e;
        lsum += e;
    }
    #pragma unroll
    for (int o = 16; o > 0; o >>= 1) lsum += __shfl_down(lsum, o, 32);
    if (lane == 0) rscal[wave] = lsum;
    __syncthreads();
    if (tid == 0) {
        float s = 0.f;
        for (int w = 0; w < 8; ++w) s += rscal[w];
        rscal[0] = 1.0f / s;
    }
    __syncthreads();
    const float inv = rscal[0];
    #pragma unroll
    for (int j = 0; j < 4; ++j) {
        int h = tid * 4 + j;
        float a = av[h] * inv;
        int s = lw[h];
        if (s >= 0) { bvec[s] = a; a = 0.0f; }  // route replaced-row weight to its step
        av[h] = a;                               // masked softmax weight
    }
    __syncthreads();

    // ---- label contribution to out: sum_t bvec[t] * x_label[t][o] ----
    const float* xcb = xc + (size_t)b * SEQ * INPUT;
    if (tid < OUTD) {
        float acc = 0.f;
        for (int t = 0; t < SEQ; ++t) acc += bvec[t] * xcb[t * INPUT + ITEM + tid];
        labp[tid] = acc;
    }
    __syncthreads();

    // ---- out = W_oh0 @ a_masked + labp : wave-per-row dot (coalesced) ----
    const float* Wob = Woh + (size_t)b * OUTD * HID;
    for (int o = wave; o < OUTD; o += 8) {
        const float* r = Wob + (size_t)o * HID;
        float acc = 0.f;
        for (int h = lane; h < HID; h += 32) acc += r[h] * av[h];
        #pragma unroll
        for (int off = 16; off > 0; off >>= 1) acc += __shfl_down(acc, off, 32);
        if (lane == 0) outp[(size_t)b * OUTD + o] = acc + labp[o];
    }

    // ---- reinst = W_reinst0^T @ a_masked + Ctx^T @ bvec : thread-per-col ----
    const float* Wrb = Wre + (size_t)b * HID * OH;
    const float* cxb = ctx + (size_t)b * SEQ * OH;
    float* rout = outp + (size_t)BATCHN * OUTD;
    if (tid < OH) {
        float acc = 0.f;
        for (int h = 0; h < HID; ++h) acc += av[h] * Wrb[(size_t)h * OH + tid];
        for (int t = 0; t < SEQ; ++t) acc += bvec[t] * cxb[t * OH + tid];
        rout[(size_t)b * OH + tid] = acc;
    }
}

// ---------------------------------------------------------------------------
extern "C" void kernel_launch(void* const* d_in, const int* in_sizes, int n_in,
                              void* d_out, int out_size, void* d_ws, size_t ws_size,
                              hipStream_t stream) {
    const float* xc  = (const float*)d_in[0];   // (128, 64, 320)
    const float* xq  = (const float*)d_in[1];   // (128, 256)
    const float* ctx = (const float*)d_in[2];   // (128, 64, 192)
    const float* Whi = (const float*)d_in[3];   // (128, 1024, 256)
    const float* Woh = (const float*)d_in[4];   // (128, 64, 1024)
    const float* Wre = (const float*)d_in[5];   // (128, 1024, 192)

    float* L0 = (float*)d_ws;                                   // 128*1024*80 f32
    float* G  = L0 + (size_t)BATCHN * HID * NPAD;               // 128*64*80  f32

    dim3 g1(9, BATCHN);   // 8 W_hi row-blocks + 1 Gram block, per batch
    k_gemm<<<g1, 256, 0, stream>>>(Whi, xc, xq, L0, G);
    k_scan<<<BATCHN, 256, 0, stream>>>(L0, G, xc, ctx, Woh, Wre, (float*)d_out);
}